// SlotAttentionTF_37529424232849
// MI455X (gfx1250) — compile-verified
//
#include <hip/hip_runtime.h>
#include <math.h>

typedef float v2f __attribute__((ext_vector_type(2)));
typedef float v8f __attribute__((ext_vector_type(8)));

// ---------------- workspace layout (float offsets) ----------------
constexpr size_t OFF_WKFG = 0;            // 256x256 folded k_fg_w@mlp_fg_w
constexpr size_t OFF_WVFG = 65536;
constexpr size_t OFF_WKBG = 131072;
constexpr size_t OFF_WVBG = 196608;
constexpr size_t OFF_GFG  = 262144;       // 4x256  grid_embed_w@mlp_fg_w
constexpr size_t OFF_GBFG = 263168;       // 256    grid_embed_b@mlp_fg_w
constexpr size_t OFF_GBG  = 263424;       // 4x256  grid_embed_w@mlp_bg_w
constexpr size_t OFF_GBBG = 264448;       // 256    grid_embed_b@mlp_bg_w
constexpr size_t OFF_MEAN = 264704;       // 16384 row means of feat
constexpr size_t OFF_RSTD = 281088;       // 16384 row rstd
constexpr size_t OFF_KM   = 297472;       // 16384x256
constexpr size_t OFF_VM   = OFF_KM + 4194304;
constexpr size_t OFF_KB   = OFF_VM + 4194304;
constexpr size_t OFF_VB   = OFF_KB + 4194304;
constexpr size_t OFF_FCN  = OFF_VB + 4194304;  // 16384x16 LN(feat_color)
constexpr size_t OFF_SFG  = OFF_FCN + 262144;  // 4x7x256 slots fg
constexpr size_t OFF_SBG  = OFF_SFG + 7168;    // 4x256   slot bg
constexpr size_t OFF_POS  = OFF_SBG + 1024;    // 4x7x2
constexpr size_t OFF_QFG  = OFF_POS + 64;      // 4x7x256
constexpr size_t OFF_QBG  = OFF_QFG + 7168;    // 4x256
constexpr size_t OFF_QC   = OFF_QBG + 1024;    // 4x7   q.gb_fg
constexpr size_t OFF_T2   = OFF_QC + 32;       // 4x7x2 (Gfg@q) folded 2-vec
constexpr size_t OFF_S    = OFF_T2 + 64;       // 4x8   attn sums
constexpr size_t OFF_P    = OFF_S + 32;        // 4x7x2 attn-weighted grid
constexpr size_t OFF_PB   = OFF_P + 64;        // 4x7x2 attn-weighted pos_bias
constexpr size_t OFF_CF   = OFF_PB + 64;       // 4x8x16 attn-weighted color
constexpr size_t OFF_UPD  = OFF_CF + 512;      // 4x8x256 partial slot updates
// end ~17.36M floats (~69.5 MB)

// out layout: slots [4,8,272] | attn [4,8,4096] | fg_position [4,7,2]
constexpr size_t OUT_ATTN = 8704;
constexpr size_t OUT_POS  = 8704 + 131072;

__device__ inline float warpSum(float v) {
#pragma unroll
  for (int m = 16; m > 0; m >>= 1) v += __shfl_xor(v, m, 32);
  return v;
}

// -------- per-row mean / rstd of feat (LN stats, fused into GEMM later) ----
__global__ void k_meanvar(const float* __restrict__ feat, float* __restrict__ mean,
                          float* __restrict__ rstd) {
  int wave = threadIdx.x >> 5, lane = threadIdx.x & 31;
  int row = blockIdx.x * 8 + wave;            // 16384 rows
  const float* f = feat + (size_t)row * 256;
  float s = 0.f, s2 = 0.f;
#pragma unroll
  for (int j = 0; j < 8; ++j) { float x = f[lane + 32 * j]; s += x; s2 += x * x; }
  s = warpSum(s); s2 = warpSum(s2);
  if (lane == 0) {
    float m = s * (1.f / 256.f);
    float v = s2 * (1.f / 256.f) - m * m;
    mean[row] = m; rstd[row] = rsqrtf(v + 1e-5f);
  }
}

// -------- LayerNorm of feat_color (16-dim rows) ---------------------------
__global__ void k_lncolor(const float* __restrict__ fc, const float* __restrict__ g,
                          const float* __restrict__ b, float* __restrict__ out) {
  int t = threadIdx.x;
  int row = blockIdx.x * 16 + (t >> 4);
  int c = t & 15;
  float x = fc[(size_t)row * 16 + c];
  float s = x, s2 = x * x;
#pragma unroll
  for (int m = 8; m > 0; m >>= 1) { s += __shfl_xor(s, m, 16); s2 += __shfl_xor(s2, m, 16); }
  float mu = s * (1.f / 16.f);
  float v = s2 * (1.f / 16.f) - mu * mu;
  out[(size_t)row * 16 + c] = (x - mu) * rsqrtf(v + 1e-5f) * g[c] + b[c];
}

// -------- tiny folded matrices: G = grid_embed_w@mlp, gb = grid_embed_b@mlp
__global__ void k_smallmat(const float* __restrict__ gew, const float* __restrict__ geb,
                           const float* __restrict__ mlpfg, const float* __restrict__ mlpbg,
                           float* __restrict__ ws) {
  int t = threadIdx.x;
  int which = blockIdx.x;                      // 0..9
  const float* mlp = (which < 5) ? mlpfg : mlpbg;
  int wl = (which < 5) ? which : which - 5;    // 0..4 (0-3: G rows, 4: gb)
  float acc = 0.f;
  for (int k = 0; k < 256; ++k) {
    float a = (wl < 4) ? gew[wl * 256 + k] : geb[k];
    acc += a * mlp[k * 256 + t];
  }
  float* dst = (which < 5) ? ((wl < 4) ? ws + OFF_GFG + wl * 256 : ws + OFF_GBFG)
                           : ((wl < 4) ? ws + OFF_GBG + wl * 256 : ws + OFF_GBBG);
  dst[t] = acc;
}

// -------- weight folding: W' = A(256x256) @ mlp(256x256), 4 products (WMMA)
__global__ void k_fold(const float* __restrict__ kfg, const float* __restrict__ vfg,
                       const float* __restrict__ kbg, const float* __restrict__ vbg,
                       const float* __restrict__ mlpfg, const float* __restrict__ mlpbg,
                       float* __restrict__ ws) {
  int wave = threadIdx.x >> 5, lane = threadIdx.x & 31;
  int lo = lane & 15, hi = lane >> 4;
  int row0 = blockIdx.x * 64 + wave * 16;
  int col0 = blockIdx.y * 16;
  int z = blockIdx.z;
  const float* A = (z == 0) ? kfg : (z == 1) ? vfg : (z == 2) ? kbg : vbg;
  const float* W = (z < 2) ? mlpfg : mlpbg;
  float* C = ws + ((z == 0) ? OFF_WKFG : (z == 1) ? OFF_WVFG : (z == 2) ? OFF_WKBG : OFF_WVBG);
  v8f acc = {};
  for (int k0 = 0; k0 < 256; k0 += 4) {
    int ca = k0 + 2 * hi;
    v2f a, bf;
    const float2 av = *(const float2*)(A + (size_t)(row0 + lo) * 256 + ca);
    a[0] = av.x; a[1] = av.y;
    bf[0] = W[(size_t)ca * 256 + col0 + lo];
    bf[1] = W[(size_t)(ca + 1) * 256 + col0 + lo];
    acc = __builtin_amdgcn_wmma_f32_16x16x4_f32(false, a, false, bf, (short)0, acc, false, false);
  }
#pragma unroll
  for (int i = 0; i < 8; ++i)
    C[(size_t)(row0 + 8 * hi + i) * 256 + col0 + lo] = acc[i];
}

// -------- main fused GEMM: LN(feat) @ {Wkfg,Wvfg,Wkbg,Wvbg} -> KM,VM,KB,VB
__global__ void k_gemm_feat(const float* __restrict__ feat, const float* __restrict__ g,
                            const float* __restrict__ bb, float* __restrict__ ws) {
  int wave = threadIdx.x >> 5, lane = threadIdx.x & 31;
  int lo = lane & 15, hi = lane >> 4;
  int row0 = blockIdx.x * 64 + wave * 16;
  int col0 = blockIdx.y * 16;
  int r = row0 + lo;
  float mu = ws[OFF_MEAN + r], rs = ws[OFF_RSTD + r];
  const float* W0 = ws + OFF_WKFG;
  const float* W1 = ws + OFF_WVFG;
  const float* W2 = ws + OFF_WKBG;
  const float* W3 = ws + OFF_WVBG;
  v8f c0 = {}, c1 = {}, c2 = {}, c3 = {};
  int cb = col0 + lo;
  for (int k0 = 0; k0 < 256; k0 += 4) {
    int ca = k0 + 2 * hi;
    const float2 fv = *(const float2*)(feat + (size_t)r * 256 + ca);
    const float2 gv = *(const float2*)(g + ca);
    const float2 bv = *(const float2*)(bb + ca);
    v2f a;
    a[0] = (fv.x - mu) * rs * gv.x + bv.x;     // LayerNorm fused into A fragment
    a[1] = (fv.y - mu) * rs * gv.y + bv.y;
    v2f b0, b1, b2, b3;
    b0[0] = W0[(size_t)ca * 256 + cb]; b0[1] = W0[(size_t)(ca + 1) * 256 + cb];
    b1[0] = W1[(size_t)ca * 256 + cb]; b1[1] = W1[(size_t)(ca + 1) * 256 + cb];
    b2[0] = W2[(size_t)ca * 256 + cb]; b2[1] = W2[(size_t)(ca + 1) * 256 + cb];
    b3[0] = W3[(size_t)ca * 256 + cb]; b3[1] = W3[(size_t)(ca + 1) * 256 + cb];
    c0 = __builtin_amdgcn_wmma_f32_16x16x4_f32(false, a, false, b0, (short)0, c0, false, false);
    c1 = __builtin_amdgcn_wmma_f32_16x16x4_f32(false, a, false, b1, (short)0, c1, false, false);
    c2 = __builtin_amdgcn_wmma_f32_16x16x4_f32(false, a, false, b2, (short)0, c2, false, false);
    c3 = __builtin_amdgcn_wmma_f32_16x16x4_f32(false, a, false, b3, (short)0, c3, false, false);
  }
  float G0 = ws[OFF_GBG + 0 * 256 + cb] - ws[OFF_GBG + 2 * 256 + cb];
  float G1 = ws[OFF_GBG + 1 * 256 + cb] - ws[OFF_GBG + 3 * 256 + cb];
  float gbv = ws[OFF_GBBG + cb];
#pragma unroll
  for (int i = 0; i < 8; ++i) {
    int rr = row0 + 8 * hi + i;
    size_t idx = (size_t)rr * 256 + cb;
    ws[OFF_KM + idx] = c0[i];
    ws[OFF_VM + idx] = c1[i];
    int n = rr & 4095;
    int y = n >> 6, x = n & 63;
    float gy = -1.f + y * (2.f / 63.f);
    float gx = -1.f + x * (2.f / 63.f);
    float add = gy * G0 + gx * G1 + gbv;       // bg grid term folded in
    ws[OFF_KB + idx] = c2[i] + add;
    ws[OFF_VB + idx] = c3[i] + add;
  }
}

// -------- init slots / positions (broadcast over batch) -------------------
__global__ void k_init(const float* __restrict__ sfg0, const float* __restrict__ sbg0,
                       const float* __restrict__ pos0, float* __restrict__ ws) {
  int t = threadIdx.x;
  int b = blockIdx.x >> 3, s = blockIdx.x & 7;
  if (s == 0) {
    ws[OFF_SBG + b * 256 + t] = sbg0[t];
  } else {
    int k = s - 1;
    ws[OFF_SFG + (size_t)(b * 7 + k) * 256 + t] = sfg0[k * 256 + t];
    if (t < 2) ws[OFF_POS + (b * 7 + k) * 2 + t] = pos0[k * 2 + t];
  }
}

// -------- q projection (LN + small GEMM) + per-slot meta ------------------
__global__ void k_q(const float* __restrict__ qlng, const float* __restrict__ qlnb,
                    const float* __restrict__ qw, const float* __restrict__ qbglng,
                    const float* __restrict__ qbglnb, const float* __restrict__ qbgw,
                    float* __restrict__ ws) {
  __shared__ float red[256], lnv[256], qsh[256];
  int t = threadIdx.x;
  int b = blockIdx.x, s = blockIdx.y;
  const float* src = (s == 0) ? ws + OFF_SBG + b * 256 : ws + OFF_SFG + (size_t)(b * 7 + s - 1) * 256;
  const float* g_ = (s == 0) ? qbglng : qlng;
  const float* b_ = (s == 0) ? qbglnb : qlnb;
  const float* W  = (s == 0) ? qbgw : qw;
  float x = src[t];
  red[t] = x; __syncthreads();
  for (int st = 128; st > 0; st >>= 1) { if (t < st) red[t] += red[t + st]; __syncthreads(); }
  float mu = red[0] * (1.f / 256.f); __syncthreads();
  float d = x - mu;
  red[t] = d * d; __syncthreads();
  for (int st = 128; st > 0; st >>= 1) { if (t < st) red[t] += red[t + st]; __syncthreads(); }
  float rs = rsqrtf(red[0] * (1.f / 256.f) + 1e-5f); __syncthreads();
  lnv[t] = d * rs * g_[t] + b_[t]; __syncthreads();
  float acc = 0.f;
  for (int k = 0; k < 256; ++k) acc += lnv[k] * W[(size_t)k * 256 + t];
  qsh[t] = acc;
  if (s == 0) ws[OFF_QBG + b * 256 + t] = acc;
  else        ws[OFF_QFG + (size_t)(b * 7 + s - 1) * 256 + t] = acc;
  __syncthreads();
  if (s > 0) {                                  // 5 dots: q.gb_fg, q.Gfg[0..3]
    for (int w2 = 0; w2 < 5; ++w2) {
      const float* vv = (w2 == 0) ? ws + OFF_GBFG : ws + OFF_GFG + (w2 - 1) * 256;
      red[t] = qsh[t] * vv[t]; __syncthreads();
      for (int st = 128; st > 0; st >>= 1) { if (t < st) red[t] += red[t + st]; __syncthreads(); }
      if (t == 0) lnv[w2] = red[0];
      __syncthreads();
    }
    if (t == 0) {
      int k = s - 1;
      ws[OFF_QC + b * 7 + k] = lnv[0];
      ws[OFF_T2 + (b * 7 + k) * 2 + 0] = lnv[1] - lnv[3];
      ws[OFF_T2 + (b * 7 + k) * 2 + 1] = lnv[2] - lnv[4];
    }
  }
}

__global__ void k_zero(float* __restrict__ ws) {
  // S | P | PB | CF | UPD are contiguous: 32+64+64+512+8192 = 8864 floats
  for (int i = threadIdx.x; i < 8864; i += 256) ws[OFF_S + i] = 0.f;
}

// -------- attention: logits, softmax over 8 slots, reductions -------------
__global__ void k_attn(const float* __restrict__ pbw, float* __restrict__ ws,
                       float* __restrict__ out, int lastIter) {
  __shared__ __align__(16) float q[8][256];
  __shared__ float qc[7], t2s[14], ps[14];
  __shared__ float sS[8], sP[14], sPB[14], sCF[128];
  int t = threadIdx.x;
  int b = blockIdx.y;
  for (int idx = t; idx < 2048; idx += 256) {
    int s = idx >> 8, d2 = idx & 255;
    q[s][d2] = (s == 0) ? ws[OFF_QBG + b * 256 + d2]
                        : ws[OFF_QFG + (size_t)(b * 7 + s - 1) * 256 + d2];
  }
  if (t < 7) qc[t] = ws[OFF_QC + b * 7 + t];
  if (t < 14) { t2s[t] = ws[OFF_T2 + b * 14 + t]; ps[t] = ws[OFF_POS + b * 14 + t]; }
  if (t < 8) sS[t] = 0.f;
  if (t < 14) { sP[t] = 0.f; sPB[t] = 0.f; }
  if (t < 128) sCF[t] = 0.f;
  __syncthreads();
  int wave = t >> 5, lane = t & 31;
  int pix = blockIdx.x * 8 + wave;
  size_t rbase = ((size_t)b * 4096 + pix) * 256;
  const float* km = ws + OFF_KM + rbase;
  const float* kb = ws + OFF_KB + rbase;
  float part[8];
#pragma unroll
  for (int s = 0; s < 8; ++s) part[s] = 0.f;
#pragma unroll
  for (int j = 0; j < 2; ++j) {
    int d0 = j * 128 + lane * 4;
    float4 kmv = *(const float4*)(km + d0);
    float4 kbv = *(const float4*)(kb + d0);
    float4 qv0 = *(const float4*)(&q[0][d0]);
    part[0] += qv0.x * kbv.x + qv0.y * kbv.y + qv0.z * kbv.z + qv0.w * kbv.w;
#pragma unroll
    for (int s = 1; s < 8; ++s) {
      float4 qv = *(const float4*)(&q[s][d0]);
      part[s] += qv.x * kmv.x + qv.y * kmv.y + qv.z * kmv.z + qv.w * kmv.w;
    }
  }
  float r[8];
#pragma unroll
  for (int s = 0; s < 8; ++s) r[s] = warpSum(part[s]);
  if (lane == 0) {
    int y = pix >> 6, x = pix & 63;
    float gy = -1.f + y * (2.f / 63.f), gx = -1.f + x * (2.f / 63.f);
    float lg[8];
    lg[0] = 0.0625f * r[0];
#pragma unroll
    for (int k = 0; k < 7; ++k)
      lg[k + 1] = 0.0625f * (r[k + 1] + qc[k] +
                             (gy - ps[2 * k]) * t2s[2 * k] + (gx - ps[2 * k + 1]) * t2s[2 * k + 1]);
    float mx = lg[0];
#pragma unroll
    for (int s = 1; s < 8; ++s) mx = fmaxf(mx, lg[s]);
    float es[8], ssum = 0.f;
#pragma unroll
    for (int s = 0; s < 8; ++s) { es[s] = __expf(lg[s] - mx); ssum += es[s]; }
    float inv = 1.f / ssum;
    float a[8];
#pragma unroll
    for (int s = 0; s < 8; ++s) {
      a[s] = es[s] * inv + 1e-8f;
      out[OUT_ATTN + ((size_t)(b * 8 + s)) * 4096 + pix] = a[s];
      atomicAdd(&sS[s], a[s]);
    }
#pragma unroll
    for (int k = 0; k < 7; ++k) {
      atomicAdd(&sP[2 * k], a[k + 1] * gy);
      atomicAdd(&sP[2 * k + 1], a[k + 1] * gx);
    }
    if (lastIter) {
      float pb0 = pbw[pix * 2], pb1 = pbw[pix * 2 + 1];
#pragma unroll
      for (int k = 0; k < 7; ++k) {
        atomicAdd(&sPB[2 * k], a[k + 1] * pb0);
        atomicAdd(&sPB[2 * k + 1], a[k + 1] * pb1);
      }
      const float* fr = ws + OFF_FCN + ((size_t)b * 4096 + pix) * 16;
#pragma unroll
      for (int s = 0; s < 8; ++s)
#pragma unroll
        for (int c = 0; c < 16; ++c) atomicAdd(&sCF[s * 16 + c], a[s] * fr[c]);
    }
  }
  __syncthreads();
  if (t < 8) atomicAdd(&ws[OFF_S + b * 8 + t], sS[t]);
  if (t < 14) {
    atomicAdd(&ws[OFF_P + b * 14 + t], sP[t]);
    if (lastIter) atomicAdd(&ws[OFF_PB + b * 14 + t], sPB[t]);
  }
  if (lastIter && t < 128) atomicAdd(&ws[OFF_CF + b * 128 + t], sCF[t]);
}

// -------- partial aw^T V reduction (n-split for parallelism) --------------
__global__ void k_upd_partial(float* __restrict__ ws, const float* __restrict__ attn) {
  int t = threadIdx.x;
  int b = blockIdx.x >> 3, s = blockIdx.x & 7;
  int n0 = blockIdx.y * 256;                  // 16 chunks of 256 pixels
  const float* arow = attn + ((size_t)(b * 8 + s)) * 4096 + n0;
  const float* V = ws + ((s == 0) ? OFF_VB : OFF_VM) + (size_t)(b * 4096 + n0) * 256;
  float acc = 0.f;
  for (int n = 0; n < 256; ++n) acc += arow[n] * V[(size_t)n * 256 + t];
  atomicAdd(&ws[OFF_UPD + (size_t)(b * 8 + s) * 256 + t], acc);
}

// -------- apply slot updates + momentum position update -------------------
__global__ void k_upd_apply(float* __restrict__ ws, const float* __restrict__ pbb,
                            int lastIter) {
  int t = threadIdx.x;
  int b = blockIdx.x >> 3, s = blockIdx.x & 7;
  float invS = 1.f / ws[OFF_S + b * 8 + s];
  float u = ws[OFF_UPD + (size_t)(b * 8 + s) * 256 + t] * invS;
  if (s == 0) {
    if (!lastIter) ws[OFF_SBG + b * 256 + t] += u;
  } else {
    int k = s - 1;
    float p0 = ws[OFF_POS + (b * 7 + k) * 2];
    float p1 = ws[OFF_POS + (b * 7 + k) * 2 + 1];
    float np0 = ws[OFF_P + b * 14 + 2 * k] * invS;
    float np1 = ws[OFF_P + b * 14 + 2 * k + 1] * invS;
    if (!lastIter) {
      float d0 = np0 - p0, d1 = np1 - p1;     // sum aw*rel folded term
      float ex = d0 * (ws[OFF_GFG + t] - ws[OFF_GFG + 512 + t]) +
                 d1 * (ws[OFF_GFG + 256 + t] - ws[OFF_GFG + 768 + t]) + ws[OFF_GBFG + t];
      ws[OFF_SFG + (size_t)(b * 7 + k) * 256 + t] += u + ex;
    }
    __syncthreads();
    if (t < 2) {
      float p = (t == 0) ? p0 : p1;
      float np = (t == 0) ? np0 : np1;
      float pn = np * 0.5f + p * 0.5f;        // MOMENTUM = 0.5
      if (lastIter)
        pn += tanhf(ws[OFF_PB + b * 14 + 2 * k + t] * invS + pbb[t]) * 0.1f;
      ws[OFF_POS + (b * 7 + k) * 2 + t] = pn;
    }
  }
}

// -------- assemble outputs -------------------------------------------------
__global__ void k_final(const float* __restrict__ ws, float* __restrict__ out) {
  int t = threadIdx.x;                         // 272
  int b = blockIdx.x, s = blockIdx.y;
  float invS = 1.f / ws[OFF_S + b * 8 + s];
  size_t obase = ((size_t)(b * 8 + s)) * 272;
  if (t < 256) {
    float v = (s == 0) ? ws[OFF_SBG + b * 256 + t]
                       : ws[OFF_SFG + (size_t)(b * 7 + s - 1) * 256 + t];
    out[obase + t] = v;
  } else {
    int c = t - 256;
    out[obase + t] = ws[OFF_CF + b * 128 + s * 16 + c] * invS;
  }
  if (s > 0 && t < 2)
    out[OUT_POS + (b * 7 + (s - 1)) * 2 + t] = ws[OFF_POS + (b * 7 + s - 1) * 2 + t];
}

extern "C" void kernel_launch(void* const* d_in, const int* in_sizes, int n_in,
                              void* d_out, int out_size, void* d_ws, size_t ws_size,
                              hipStream_t stream) {
  (void)in_sizes; (void)n_in; (void)out_size; (void)ws_size;
  const float* feat       = (const float*)d_in[0];
  const float* feat_color = (const float*)d_in[1];
  const float* gew        = (const float*)d_in[2];
  const float* geb        = (const float*)d_in[3];
  const float* kfg        = (const float*)d_in[4];
  const float* vfg        = (const float*)d_in[5];
  const float* kbg        = (const float*)d_in[6];
  const float* vbg        = (const float*)d_in[7];
  const float* mlpfg      = (const float*)d_in[8];
  const float* mlpbg      = (const float*)d_in[9];
  const float* pos0       = (const float*)d_in[10];
  const float* sfg0       = (const float*)d_in[11];
  const float* sbg0       = (const float*)d_in[12];
  const float* pbw        = (const float*)d_in[13];
  const float* pbb        = (const float*)d_in[14];
  const float* qlng       = (const float*)d_in[15];
  const float* qlnb       = (const float*)d_in[16];
  const float* qw         = (const float*)d_in[17];
  const float* qbglng     = (const float*)d_in[18];
  const float* qbglnb     = (const float*)d_in[19];
  const float* qbgw       = (const float*)d_in[20];
  const float* nfg        = (const float*)d_in[21];
  const float* nfb        = (const float*)d_in[22];
  const float* ncg        = (const float*)d_in[23];
  const float* ncb        = (const float*)d_in[24];
  float* out = (float*)d_out;
  float* ws  = (float*)d_ws;

  k_meanvar<<<2048, 256, 0, stream>>>(feat, ws + OFF_MEAN, ws + OFF_RSTD);
  k_lncolor<<<1024, 256, 0, stream>>>(feat_color, ncg, ncb, ws + OFF_FCN);
  k_smallmat<<<10, 256, 0, stream>>>(gew, geb, mlpfg, mlpbg, ws);
  k_fold<<<dim3(4, 16, 4), 128, 0, stream>>>(kfg, vfg, kbg, vbg, mlpfg, mlpbg, ws);
  k_gemm_feat<<<dim3(256, 16), 128, 0, stream>>>(feat, nfg, nfb, ws);
  k_init<<<32, 256, 0, stream>>>(sfg0, sbg0, pos0, ws);
  for (int it = 0; it < 4; ++it) {
    int last = (it == 3) ? 1 : 0;
    k_q<<<dim3(4, 8), 256, 0, stream>>>(qlng, qlnb, qw, qbglng, qbglnb, qbgw, ws);
    k_zero<<<1, 256, 0, stream>>>(ws);
    k_attn<<<dim3(512, 4), 256, 0, stream>>>(pbw, ws, out, last);
    if (!last) k_upd_partial<<<dim3(32, 16), 256, 0, stream>>>(ws, out + OUT_ATTN);
    k_upd_apply<<<32, 256, 0, stream>>>(ws, pbb, last);
  }
  k_final<<<dim3(4, 8), 272, 0, stream>>>(ws, out);
}